// AGCRNCell_44238163148928
// MI455X (gfx1250) — compile-verified
//
#include <hip/hip_runtime.h>

typedef float v2f __attribute__((ext_vector_type(2)));
typedef float v8f __attribute__((ext_vector_type(8)));
typedef unsigned int  u32;
typedef unsigned long long u64;
typedef u32 v4u __attribute__((ext_vector_type(4)));
typedef int  v4i __attribute__((ext_vector_type(4)));
typedef int  v8i __attribute__((ext_vector_type(8)));

// Problem constants (from reference setup_inputs)
#define NB   64          // batch
#define NN   2000        // nodes
#define CIN  128         // Din + Dout concat
#define BC   (NB * CIN)  // 8192 columns of the diffusion GEMM
#define ED   16          // embedding dim
#define HD   64          // hidden dim

// ---------------------------------------------------------------------------
// TDM: DMA a 2D tile (tileD1 rows x tileD0 dwords) from global into LDS,
// inserting 1 pad dword after every 2^(padCode+1) dwords (== after each row).
// remD0/remD1 = valid extent from tile start (OOB reads return zero).
// D# bit layout per CDNA5 ISA ch.8 (Group0 128b, Group1 256b).
// ---------------------------------------------------------------------------
__device__ __forceinline__ void tdm_load_tile(u32 ldsOff, const float* gtile,
                                              u32 tileD0, u32 tileD1,
                                              u32 remD0, u32 remD1,
                                              u32 strideD0, u32 padCode) {
  const u64 ga = (u64)(size_t)gtile;
  v4u g0;
  g0.x = 1u;                                              // count=1, flags=0
  g0.y = ldsOff;                                          // lds_addr [63:32]
  g0.z = (u32)ga;                                         // global_addr lo
  g0.w = ((u32)(ga >> 32) & 0x01ffffffu) | (2u << 30);    // addr hi | type=2
  v8i g1;
  g1[0] = (int)((2u << 16) | (1u << 20) | (padCode << 22)); // 4B elems, pad_en, pad_interval, pad_amount=0(=1 dword)
  g1[1] = (int)((remD0 & 0xffffu) << 16);                   // tensor_dim0 lo16
  g1[2] = (int)((remD0 >> 16) | ((remD1 & 0xffffu) << 16)); // dim0 hi | dim1 lo
  g1[3] = (int)((remD1 >> 16) | (tileD0 << 16));            // dim1 hi | tile_dim0
  g1[4] = (int)tileD1;                                      // tile_dim1 (tile_dim2=0)
  g1[5] = (int)strideD0;                                    // tensor_dim0_stride lo32
  g1[6] = 0;                                                // stride0 hi | stride1 lo
  g1[7] = 0;
  v4i gz = {0, 0, 0, 0};
#if defined(__clang_major__) && __clang_major__ >= 23
  v8i gz8 = {0, 0, 0, 0, 0, 0, 0, 0};
  __builtin_amdgcn_tensor_load_to_lds(g0, g1, gz, gz, gz8, 0);
#else
  __builtin_amdgcn_tensor_load_to_lds(g0, g1, gz, gz, 0);
#endif
}

// ---------------------------------------------------------------------------
// 1) A = softmax(relu(E @ E^T), axis=1)    (one block per row)
// ---------------------------------------------------------------------------
__global__ __launch_bounds__(256) void compute_adj(const float* __restrict__ E,
                                                   float* __restrict__ Aout) {
  const int row = blockIdx.x;
  __shared__ float er[ED];
  __shared__ float red[256];
  const int tid = threadIdx.x;
  if (tid < ED) er[tid] = E[row * ED + tid];
  __syncthreads();

  float lmax = -1e30f;
  for (int c = tid; c < NN; c += 256) {
    float s = 0.f;
#pragma unroll
    for (int d = 0; d < ED; ++d) s += er[d] * E[c * ED + d];
    s = fmaxf(s, 0.f);
    Aout[(size_t)row * NN + c] = s;
    lmax = fmaxf(lmax, s);
  }
  red[tid] = lmax;
  __syncthreads();
  for (int off = 128; off > 0; off >>= 1) {
    if (tid < off) red[tid] = fmaxf(red[tid], red[tid + off]);
    __syncthreads();
  }
  const float m = red[0];
  __syncthreads();

  float lsum = 0.f;
  for (int c = tid; c < NN; c += 256) {
    float e = __expf(Aout[(size_t)row * NN + c] - m);
    Aout[(size_t)row * NN + c] = e;
    lsum += e;
  }
  red[tid] = lsum;
  __syncthreads();
  for (int off = 128; off > 0; off >>= 1) {
    if (tid < off) red[tid] += red[tid + off];
    __syncthreads();
  }
  const float inv = 1.f / red[0];
  __syncthreads();
  for (int c = tid; c < NN; c += 256) Aout[(size_t)row * NN + c] *= inv;
}

// ---------------------------------------------------------------------------
// 2) Pack Xc[m, b*128 + c] = (c<64) ? x[b,m,c] : state[b,m,c-64]
// ---------------------------------------------------------------------------
__global__ __launch_bounds__(256) void build_xc(const float* __restrict__ x,
                                                const float* __restrict__ state,
                                                float* __restrict__ Xc) {
  const size_t total = (size_t)NN * BC;
  for (size_t o = (size_t)blockIdx.x * blockDim.x + threadIdx.x; o < total;
       o += (size_t)gridDim.x * blockDim.x) {
    const int c = (int)(o & 127);
    const size_t t = o >> 7;          // = m*64 + b
    const int b = (int)(t & 63);
    const int m = (int)(t >> 6);
    float v;
    if (c < HD) v = x[((size_t)b * NN + m) * HD + c];
    else        v = state[((size_t)b * NN + m) * HD + (c - HD)];
    Xc[o] = v;
  }
}

// ---------------------------------------------------------------------------
// 3) Per-node weights: out[n, j] = sum_d E[n,d] * Wp[d, j]   (J cols)
// ---------------------------------------------------------------------------
__global__ __launch_bounds__(256) void node_weights(const float* __restrict__ E,
                                                    const float* __restrict__ Wp,
                                                    float* __restrict__ out, int J) {
  const int n = blockIdx.y;
  const int j = blockIdx.x * 256 + threadIdx.x;
  if (j >= J) return;
  float acc = 0.f;
#pragma unroll
  for (int d = 0; d < ED; ++d) acc += E[n * ED + d] * Wp[(size_t)d * J + j];
  out[(size_t)n * J + j] = acc;
}

// ---------------------------------------------------------------------------
// 4) fp32 WMMA GEMM: C[M,N] = A[M,K] @ B[K,N]   (K % 16 == 0, N % 64 == 0)
//    Block tile 128x64, 8 waves; double-buffered LDS tiles filled by the
//    Tensor Data Mover (wave 0 issues, TENSORcnt + barrier publishes).
// ---------------------------------------------------------------------------
#define TM 128
#define TN 64
#define TK 16

__global__ __launch_bounds__(256) void gemm_wmma_f32(const float* __restrict__ A,
                                                     const float* __restrict__ B,
                                                     float* __restrict__ C,
                                                     int M, int K, int N) {
  __shared__ float As[2][TM][TK + 1];  // stride 17 == TDM pad(16dw interval,1dw)
  __shared__ float Bs[2][TK][TN + 1];  // stride 65 == TDM pad(64dw interval,1dw)

  const int tid  = threadIdx.x;
  const int wave = tid >> 5;
  const int lane = tid & 31;
  const int half = lane >> 4;
  const int l    = lane & 15;
  const int m0   = blockIdx.y * TM;
  const int n0   = blockIdx.x * TN;

  const u32 asBase  = (u32)(size_t)(&As[0][0][0]);  // low 32 bits = LDS offset
  const u32 bsBase  = (u32)(size_t)(&Bs[0][0][0]);
  const u32 asBytes = (u32)sizeof(As[0]);
  const u32 bsBytes = (u32)sizeof(Bs[0]);

  v8f acc[4] = {};
  const int kTiles = K / TK;

  auto issue = [&](int kt, int buf) {
    const int kb = kt * TK;
    // A tile: TM rows x TK dwords, row stride K; zero-fill rows past M.
    tdm_load_tile(asBase + (u32)buf * asBytes, A + (size_t)m0 * K + kb,
                  TK, TM, (u32)(K - kb), (u32)(M - m0), (u32)K, 3u);
    // B tile: TK rows x TN dwords, row stride N.
    tdm_load_tile(bsBase + (u32)buf * bsBytes, B + (size_t)kb * N + n0,
                  TN, TK, (u32)(N - n0), (u32)(K - kb), (u32)N, 5u);
  };

  if (wave == 0) {
    issue(0, 0);
    __builtin_amdgcn_s_wait_tensorcnt(0);
  }
  __syncthreads();

  for (int kt = 0; kt < kTiles; ++kt) {
    const int cur = kt & 1;
    if (wave == 0 && kt + 1 < kTiles) issue(kt + 1, cur ^ 1);  // prefetch next

#pragma unroll
    for (int kk = 0; kk < 4; ++kk) {
      const int ak = kk * 4 + half * 2;  // lanes 0-15: K0/K1, lanes 16-31: K2/K3
      v2f a;
      a.x = As[cur][wave * 16 + l][ak];
      a.y = As[cur][wave * 16 + l][ak + 1];
#pragma unroll
      for (int nt = 0; nt < 4; ++nt) {
        v2f b;
        b.x = Bs[cur][ak][nt * 16 + l];
        b.y = Bs[cur][ak + 1][nt * 16 + l];
        acc[nt] = __builtin_amdgcn_wmma_f32_16x16x4_f32(
            false, a, false, b, (short)0, acc[nt], false, false);
      }
    }

    if (wave == 0 && kt + 1 < kTiles) __builtin_amdgcn_s_wait_tensorcnt(0);
    __syncthreads();
  }

  // C/D layout: VGPR r: lanes0-15 -> M=r, lanes16-31 -> M=8+r; N = lane&15
#pragma unroll
  for (int nt = 0; nt < 4; ++nt) {
#pragma unroll
    for (int r = 0; r < 8; ++r) {
      const int gm = m0 + wave * 16 + half * 8 + r;
      const int gn = n0 + nt * 16 + l;
      if (gm < M) C[(size_t)gm * N + gn] = acc[nt][r];
    }
  }
}

// ---------------------------------------------------------------------------
// 5) Gate projection: z_r = sigmoid(xg @ Wg[n] + bias[n]);
//    writes r -> R, and z*state in-place into Xc second half (candidate).
// ---------------------------------------------------------------------------
__global__ __launch_bounds__(256) void gate_proj(const float* __restrict__ E,
                                                 const float* __restrict__ gate_b,
                                                 const float* __restrict__ Wg,
                                                 const float* __restrict__ T1,
                                                 const float* __restrict__ T2,
                                                 float* __restrict__ Xc,
                                                 const float* __restrict__ state,
                                                 float* __restrict__ R) {
  const int n = blockIdx.x;
  __shared__ float xs[16][385];  // 16 batches x 384 (k*128+i)
  __shared__ float bias[128];
  const int tid = threadIdx.x;

  if (tid < 128) {
    float a = 0.f;
#pragma unroll
    for (int d = 0; d < ED; ++d) a += E[n * ED + d] * gate_b[d * 128 + tid];
    bias[tid] = a;
  }
  const float* xrow  = Xc + (size_t)n * BC;
  const float* t1row = T1 + (size_t)n * BC;
  const float* t2row = T2 + (size_t)n * BC;
  const float* wrow  = Wg + (size_t)n * (3 * CIN * 128);

  for (int bt = 0; bt < 4; ++bt) {
    __syncthreads();
    for (int idx = tid; idx < 16 * 384; idx += 256) {
      const int bb = idx / 384, j = idx % 384;
      const int k = j >> 7, i = j & 127;
      const int b = bt * 16 + bb;
      const size_t col = (size_t)b * CIN + i;
      float v;
      if (k == 0)      v = xrow[col];
      else if (k == 1) v = t1row[col];
      else             v = 2.f * t2row[col] - xrow[col];  // S2 = 2A^2 - I
      xs[bb][j] = v;
    }
    __syncthreads();
#pragma unroll
    for (int e = 0; e < 8; ++e) {
      const int lin = tid + e * 256;  // 0..2047
      const int bb = lin >> 7, o = lin & 127;
      float acc = bias[o];
      const float* xv = xs[bb];
      for (int j = 0; j < 384; ++j) acc += xv[j] * wrow[(size_t)j * 128 + o];
      const float zr = 1.f / (1.f + __expf(-acc));
      const int b = bt * 16 + bb;
      if (o < HD) {  // z: candidate second half = z * state
        const float st = state[((size_t)b * NN + n) * HD + o];
        Xc[(size_t)n * BC + (size_t)b * CIN + HD + o] = zr * st;
      } else {       // r
        R[((size_t)b * NN + n) * HD + (o - HD)] = zr;
      }
    }
  }
}

// ---------------------------------------------------------------------------
// 6) Update projection: hc = tanh(xg @ Wu[n] + bias); h = r*state + (1-r)*hc
// ---------------------------------------------------------------------------
__global__ __launch_bounds__(256) void upd_proj(const float* __restrict__ E,
                                                const float* __restrict__ upd_b,
                                                const float* __restrict__ Wu,
                                                const float* __restrict__ T1,
                                                const float* __restrict__ T2,
                                                const float* __restrict__ Xc,
                                                const float* __restrict__ state,
                                                const float* __restrict__ R,
                                                float* __restrict__ H) {
  const int n = blockIdx.x;
  __shared__ float xs[16][385];
  __shared__ float bias[64];
  const int tid = threadIdx.x;

  if (tid < 64) {
    float a = 0.f;
#pragma unroll
    for (int d = 0; d < ED; ++d) a += E[n * ED + d] * upd_b[d * 64 + tid];
    bias[tid] = a;
  }
  const float* xrow  = Xc + (size_t)n * BC;
  const float* t1row = T1 + (size_t)n * BC;
  const float* t2row = T2 + (size_t)n * BC;
  const float* wrow  = Wu + (size_t)n * (3 * CIN * HD);

  for (int bt = 0; bt < 4; ++bt) {
    __syncthreads();
    for (int idx = tid; idx < 16 * 384; idx += 256) {
      const int bb = idx / 384, j = idx % 384;
      const int k = j >> 7, i = j & 127;
      const int b = bt * 16 + bb;
      const size_t col = (size_t)b * CIN + i;
      float v;
      if (k == 0)      v = xrow[col];
      else if (k == 1) v = t1row[col];
      else             v = 2.f * t2row[col] - xrow[col];
      xs[bb][j] = v;
    }
    __syncthreads();
#pragma unroll
    for (int e = 0; e < 4; ++e) {
      const int lin = tid + e * 256;  // 0..1023
      const int bb = lin >> 6, o = lin & 63;
      float acc = bias[o];
      const float* xv = xs[bb];
      for (int j = 0; j < 384; ++j) acc += xv[j] * wrow[(size_t)j * HD + o];
      const float hc = tanhf(acc);
      const int b = bt * 16 + bb;
      const size_t oi = ((size_t)b * NN + n) * HD + o;
      const float r = R[oi];
      H[oi] = r * state[oi] + (1.f - r) * hc;
    }
  }
}

// ---------------------------------------------------------------------------
// Launcher
// ---------------------------------------------------------------------------
extern "C" void kernel_launch(void* const* d_in, const int* in_sizes, int n_in,
                              void* d_out, int out_size, void* d_ws, size_t ws_size,
                              hipStream_t stream) {
  const float* x      = (const float*)d_in[0];
  const float* state  = (const float*)d_in[1];
  const float* E      = (const float*)d_in[2];
  const float* gate_W = (const float*)d_in[3];
  const float* gate_b = (const float*)d_in[4];
  const float* upd_W  = (const float*)d_in[5];
  const float* upd_b  = (const float*)d_in[6];

  float* H    = (float*)d_out;                          // [B,N,64]
  float* Wout = (float*)d_out + (size_t)NB * NN * HD;   // [N,3,128,64] = weights

  float* ws = (float*)d_ws;
  float* Adj = ws;                                      //  4,000,000
  float* Xc  = Adj + (size_t)NN * NN;                   // 16,384,000
  float* T1  = Xc + (size_t)NN * BC;                    // 16,384,000
  float* T2  = T1 + (size_t)NN * BC;                    // 16,384,000
  float* Wg  = T2 + (size_t)NN * BC;                    // 98,304,000
  float* R   = Wg + (size_t)NN * 3 * CIN * 128;         //  8,192,000

  // 1) adjacency
  compute_adj<<<NN, 256, 0, stream>>>(E, Adj);
  // 2) pack [x || state]
  build_xc<<<(int)(((size_t)NN * BC) / 256), 256, 0, stream>>>(x, state, Xc);
  // 3) per-node weights (upd goes straight to output region 2)
  node_weights<<<dim3((3 * CIN * 128) / 256, NN), 256, 0, stream>>>(E, gate_W, Wg, 3 * CIN * 128);
  node_weights<<<dim3((3 * CIN * HD) / 256, NN), 256, 0, stream>>>(E, upd_W, Wout, 3 * CIN * HD);

  dim3 gg(BC / TN, (NN + TM - 1) / TM);
  // 4) gate diffusion
  gemm_wmma_f32<<<gg, 256, 0, stream>>>(Adj, Xc, T1, NN, NN, BC);
  gemm_wmma_f32<<<gg, 256, 0, stream>>>(Adj, T1, T2, NN, NN, BC);
  // 5) gate projection (+ candidate in place)
  gate_proj<<<NN, 256, 0, stream>>>(E, gate_b, Wg, T1, T2, Xc, state, R);
  // 6) candidate diffusion
  gemm_wmma_f32<<<gg, 256, 0, stream>>>(Adj, Xc, T1, NN, NN, BC);
  gemm_wmma_f32<<<gg, 256, 0, stream>>>(Adj, T1, T2, NN, NN, BC);
  // 7) update projection -> h
  upd_proj<<<NN, 256, 0, stream>>>(E, upd_b, Wout, T1, T2, Xc, state, R, H);
}